// HANLayer_5729486373121
// MI455X (gfx1250) — compile-verified
//
#include <hip/hip_runtime.h>
#include <cstdint>
#include <cstddef>

// ---------------- problem constants (fixed by the reference) ----------------
#define NODES   50000
#define IN_DIMS 128
#define OUT_DIMS 32
#define NHEADS  8
#define NEDGE   800000
#define FDIM    256     // OUT_DIMS * NHEADS
#define SEMHID  128

typedef __attribute__((ext_vector_type(2))) float v2f;
typedef __attribute__((ext_vector_type(8))) float v8f;

// Order-preserving float<->uint encoding for atomicMax-based segment max.
// memset(0) == -inf in this encoding.
__device__ __forceinline__ unsigned f2ord(float f) {
    unsigned u = __float_as_uint(f);
    return (u & 0x80000000u) ? ~u : (u | 0x80000000u);
}
__device__ __forceinline__ float ord2f(unsigned u) {
    return (u & 0x80000000u) ? __uint_as_float(u & 0x7FFFFFFFu)
                             : __uint_as_float(~u);
}

// ---------------------------------------------------------------------------
// Kernel 1: C[M x NOUT] = A[M x K] @ B[K x NOUT], fp32 via V_WMMA_F32_16X16X4_F32
// block = (32, 8): 8 waves, each wave owns one 16x16 C tile.
// grid  = (M/16, NOUT/(16*8))
// WMMA f32 A-frag layout (16x4): lanes 0-15 = rows, VGPR0={K0,K2}, VGPR1={K1,K3}
// (lanes 16-31 carry K+2). B (4x16): row K striped across lanes within VGPR.
// C (16x16): VGPR r = rows {r, r+8} split across lane halves.
// ---------------------------------------------------------------------------
__global__ __launch_bounds__(256) void wmma_gemm_f32(
    const float* __restrict__ A, const float* __restrict__ B,
    float* __restrict__ C, int K, int NOUT)
{
    const int lane = threadIdx.x;            // 0..31
    const int wave = threadIdx.y;            // 0..7
    const int m0   = blockIdx.x * 16;
    const int n0   = (blockIdx.y * 8 + wave) * 16;
    if (n0 >= NOUT) return;
    const int half = lane >> 4;              // 0: K+0/K+1 ; 1: K+2/K+3
    const int l    = lane & 15;

    v8f acc = {};
    const float* Arow = A + (size_t)(m0 + l) * K;
    for (int k = 0; k < K; k += 4) {
        const int ka = k + 2 * half;
        v2f a, b;
        a.x = Arow[ka];
        a.y = Arow[ka + 1];
        b.x = B[(size_t)ka       * NOUT + n0 + l];
        b.y = B[(size_t)(ka + 1) * NOUT + n0 + l];
        acc = __builtin_amdgcn_wmma_f32_16x16x4_f32(
            false, a, false, b, (short)0, acc, false, false);
    }
    float* Crow = C + (size_t)(m0 + half * 8) * NOUT + n0 + l;
#pragma unroll
    for (int r = 0; r < 8; ++r)
        Crow[(size_t)r * NOUT] = acc[r];
}

// ---------------------------------------------------------------------------
// Kernel 2: per-(node,head) attention scores el/er = sum_d feat*al / feat*ar
// ---------------------------------------------------------------------------
__global__ void scores_kernel(const float* __restrict__ feat,
                              const float* __restrict__ al,
                              const float* __restrict__ ar,
                              float* __restrict__ el, float* __restrict__ er)
{
    int idx = blockIdx.x * blockDim.x + threadIdx.x;   // node*NHEADS + head
    if (idx >= NODES * NHEADS) return;
    const int n = idx / NHEADS, hh = idx % NHEADS;
    const float* f   = feat + (size_t)n * FDIM + hh * OUT_DIMS;
    const float* alh = al + hh * OUT_DIMS;
    const float* arh = ar + hh * OUT_DIMS;
    float sl = 0.f, sr = 0.f;
#pragma unroll 8
    for (int d = 0; d < OUT_DIMS; ++d) { float v = f[d]; sl += v * alh[d]; sr += v * arh[d]; }
    el[idx] = sl;
    er[idx] = sr;
}

// ---------------------------------------------------------------------------
// Kernel 3: per-(edge,head) leaky_relu score -> atomic segment max over dst
// ---------------------------------------------------------------------------
__global__ void edge_max_kernel(const int* __restrict__ src, const int* __restrict__ dst,
                                const float* __restrict__ el, const float* __restrict__ er,
                                unsigned* __restrict__ m)
{
    long idx = (long)blockIdx.x * blockDim.x + threadIdx.x;   // edge*NHEADS + head
    if (idx >= (long)NEDGE * NHEADS) return;
    const int e = (int)(idx / NHEADS), hh = (int)(idx % NHEADS);
    const int s = src[e], d = dst[e];
    float x = el[s * NHEADS + hh] + er[d * NHEADS + hh];
    x = x > 0.f ? x : 0.2f * x;
    atomicMax(&m[d * NHEADS + hh], f2ord(x));
}

// ---------------------------------------------------------------------------
// Kernel 4: one wave32 per edge. Lanes 0-7 compute ex per head + denom atomics,
// then all 32 lanes scatter feat[src]*ex into acc[dst] (8 heads x 32 dims).
// ---------------------------------------------------------------------------
__global__ __launch_bounds__(256) void edge_scatter_kernel(
    const int* __restrict__ src, const int* __restrict__ dst,
    const float* __restrict__ el, const float* __restrict__ er,
    const unsigned* __restrict__ m, const float* __restrict__ feat,
    float* __restrict__ denom, float* __restrict__ acc)
{
    const int lane = threadIdx.x & 31;
    const int wid  = threadIdx.x >> 5;
    const long e = (long)blockIdx.x * 8 + wid;
    if (e >= NEDGE) return;
    const int s = src[e], d = dst[e];

    float ex = 0.f;
    if (lane < NHEADS) {
        float x = el[s * NHEADS + lane] + er[d * NHEADS + lane];
        x = x > 0.f ? x : 0.2f * x;
        ex = expf(x - ord2f(m[d * NHEADS + lane]));
        atomicAdd(&denom[d * NHEADS + lane], ex);
    }
    const float* fs = feat + (size_t)s * FDIM;
    float*       ad = acc  + (size_t)d * FDIM;
#pragma unroll
    for (int j = 0; j < NHEADS; ++j) {
        const float w = __shfl(ex, j, 32);
        const int f = j * OUT_DIMS + lane;         // coalesced 128B per head
        atomicAdd(&ad[f], fs[f] * w);
    }
}

// ---------------------------------------------------------------------------
// Kernel 5: normalize by denom, add bias, ELU (in place: acc -> z)
// ---------------------------------------------------------------------------
__global__ void finalize_z_kernel(const float* __restrict__ denom,
                                  const float* __restrict__ bias,
                                  float* __restrict__ acc)
{
    size_t idx = (size_t)blockIdx.x * blockDim.x + threadIdx.x;
    if (idx >= (size_t)NODES * FDIM) return;
    const int n = (int)(idx / FDIM), f = (int)(idx % FDIM), hh = f / OUT_DIMS;
    const float dn = denom[n * NHEADS + hh];
    float v = acc[idx] / (dn > 0.f ? dn : 1.f) + bias[f];
    acc[idx] = v > 0.f ? v : (expf(v) - 1.f);
}

// ---------------------------------------------------------------------------
// Kernel 6: semantic attention logits, fused WMMA GEMM + tanh epilogue:
// q[n] += sum_col tanh( (z @ Wsem)[n,col] + bsem[col] ) * wsem[col]
// block = (32, 8): wave owns one 16x16 hidden tile, SEMHID=128 = 8 tiles.
// ---------------------------------------------------------------------------
__global__ __launch_bounds__(256) void sem_q_kernel(
    const float* __restrict__ Z, const float* __restrict__ Wsem,
    const float* __restrict__ bsem, const float* __restrict__ wsem,
    float* __restrict__ q)
{
    const int lane = threadIdx.x, wave = threadIdx.y;
    const int m0 = blockIdx.x * 16;
    const int n0 = wave * 16;
    const int half = lane >> 4, l = lane & 15;

    v8f acc = {};
    const float* Zrow = Z + (size_t)(m0 + l) * FDIM;
    for (int k = 0; k < FDIM; k += 4) {
        const int ka = k + 2 * half;
        v2f a, b;
        a.x = Zrow[ka];
        a.y = Zrow[ka + 1];
        b.x = Wsem[(size_t)ka       * SEMHID + n0 + l];
        b.y = Wsem[(size_t)(ka + 1) * SEMHID + n0 + l];
        acc = __builtin_amdgcn_wmma_f32_16x16x4_f32(
            false, a, false, b, (short)0, acc, false, false);
    }
    const float bs = bsem[n0 + l];
    const float wv = wsem[n0 + l];
#pragma unroll
    for (int r = 0; r < 8; ++r) {
        float p = tanhf(acc[r] + bs) * wv;
        // reduce across the 16 lanes of each half (rows r and r+8)
        for (int off = 8; off > 0; off >>= 1)
            p += __shfl_xor(p, off, 16);
        if (l == 0)
            atomicAdd(&q[m0 + half * 8 + r], p);
    }
}

// ---------------------------------------------------------------------------
// Kernel 7: block-reduce q_a / q_b into global sums[2]
// ---------------------------------------------------------------------------
__global__ __launch_bounds__(256) void qsum_kernel(const float* __restrict__ qa,
                                                   const float* __restrict__ qb,
                                                   float* __restrict__ sums)
{
    __shared__ float sa[256], sb[256];
    const int tid = threadIdx.x;
    const size_t i = (size_t)blockIdx.x * 256 + tid;
    sa[tid] = (i < NODES) ? qa[i] : 0.f;
    sb[tid] = (i < NODES) ? qb[i] : 0.f;
    __syncthreads();
    for (int off = 128; off > 0; off >>= 1) {
        if (tid < off) { sa[tid] += sa[tid + off]; sb[tid] += sb[tid + off]; }
        __syncthreads();
    }
    if (tid == 0) { atomicAdd(&sums[0], sa[0]); atomicAdd(&sums[1], sb[0]); }
}

// ---------------------------------------------------------------------------
// Kernel 8: beta = softmax(mean q); out = beta0*z_a + beta1*z_b
// ---------------------------------------------------------------------------
__global__ void combine_kernel(const float* __restrict__ za,
                               const float* __restrict__ zb,
                               const float* __restrict__ sums,
                               float* __restrict__ out)
{
    size_t idx = (size_t)blockIdx.x * blockDim.x + threadIdx.x;
    if (idx >= (size_t)NODES * FDIM) return;
    const float q0 = sums[0] * (1.0f / NODES);
    const float q1 = sums[1] * (1.0f / NODES);
    const float mx = fmaxf(q0, q1);
    const float e0 = expf(q0 - mx), e1 = expf(q1 - mx);
    const float inv = 1.0f / (e0 + e1);
    out[idx] = (e0 * inv) * za[idx] + (e1 * inv) * zb[idx];
}

// ---------------------------------------------------------------------------
extern "C" void kernel_launch(void* const* d_in, const int* in_sizes, int n_in,
                              void* d_out, int out_size, void* d_ws, size_t ws_size,
                              hipStream_t stream)
{
    const float* h      = (const float*)d_in[0];
    const int*   src_a  = (const int*)  d_in[1];
    const int*   dst_a  = (const int*)  d_in[2];
    const int*   src_b  = (const int*)  d_in[3];
    const int*   dst_b  = (const int*)  d_in[4];
    const float* W_a    = (const float*)d_in[5];
    const float* al_a   = (const float*)d_in[6];
    const float* ar_a   = (const float*)d_in[7];
    const float* bias_a = (const float*)d_in[8];
    const float* W_b    = (const float*)d_in[9];
    const float* al_b   = (const float*)d_in[10];
    const float* ar_b   = (const float*)d_in[11];
    const float* bias_b = (const float*)d_in[12];
    const float* Wsem   = (const float*)d_in[13];
    const float* bsem   = (const float*)d_in[14];
    const float* wsem   = (const float*)d_in[15];
    float* out = (float*)d_out;

    // ---- workspace carve-up (256B aligned) ----
    char* ws = (char*)d_ws;
    size_t off = 0;
    auto alloc = [&](size_t bytes) -> void* {
        void* p = ws + off;
        off += (bytes + 255) & ~(size_t)255;
        return p;
    };
    float*    feat_a  = (float*)   alloc((size_t)NODES * FDIM * 4);
    float*    feat_b  = (float*)   alloc((size_t)NODES * FDIM * 4);
    float*    acc_a   = (float*)   alloc((size_t)NODES * FDIM * 4);   // -> z_a
    float*    acc_b   = (float*)   alloc((size_t)NODES * FDIM * 4);   // -> z_b
    float*    el_a    = (float*)   alloc((size_t)NODES * NHEADS * 4);
    float*    er_a    = (float*)   alloc((size_t)NODES * NHEADS * 4);
    float*    el_b    = (float*)   alloc((size_t)NODES * NHEADS * 4);
    float*    er_b    = (float*)   alloc((size_t)NODES * NHEADS * 4);
    unsigned* m_a     = (unsigned*)alloc((size_t)NODES * NHEADS * 4);
    unsigned* m_b     = (unsigned*)alloc((size_t)NODES * NHEADS * 4);
    float*    den_a   = (float*)   alloc((size_t)NODES * NHEADS * 4);
    float*    den_b   = (float*)   alloc((size_t)NODES * NHEADS * 4);
    float*    q_a     = (float*)   alloc((size_t)NODES * 4);
    float*    q_b     = (float*)   alloc((size_t)NODES * 4);
    float*    sums    = (float*)   alloc(2 * 4);

    // ---- per-call zero init of accumulation buffers (graph-capture safe) ----
    hipMemsetAsync(acc_a, 0, (size_t)NODES * FDIM * 4, stream);
    hipMemsetAsync(acc_b, 0, (size_t)NODES * FDIM * 4, stream);
    hipMemsetAsync(m_a,   0, (size_t)NODES * NHEADS * 4, stream);  // ordered -inf
    hipMemsetAsync(m_b,   0, (size_t)NODES * NHEADS * 4, stream);
    hipMemsetAsync(den_a, 0, (size_t)NODES * NHEADS * 4, stream);
    hipMemsetAsync(den_b, 0, (size_t)NODES * NHEADS * 4, stream);
    hipMemsetAsync(q_a,   0, (size_t)NODES * 4, stream);
    hipMemsetAsync(q_b,   0, (size_t)NODES * 4, stream);
    hipMemsetAsync(sums,  0, 2 * 4, stream);

    const dim3 wblk(32, 8);                       // 8 wave32s per block

    // 1) feature GEMMs (WMMA):  feat = h @ W  [50000x128 @ 128x256]
    wmma_gemm_f32<<<dim3(NODES / 16, FDIM / 128), wblk, 0, stream>>>(h, W_a, feat_a, IN_DIMS, FDIM);
    wmma_gemm_f32<<<dim3(NODES / 16, FDIM / 128), wblk, 0, stream>>>(h, W_b, feat_b, IN_DIMS, FDIM);

    // 2) attention scores
    {
        const int n = NODES * NHEADS, blk = 256, grd = (n + blk - 1) / blk;
        scores_kernel<<<grd, blk, 0, stream>>>(feat_a, al_a, ar_a, el_a, er_a);
        scores_kernel<<<grd, blk, 0, stream>>>(feat_b, al_b, ar_b, el_b, er_b);
    }

    // 3) segment max over dst (softmax stabilization)
    {
        const long n = (long)NEDGE * NHEADS; const int blk = 256;
        const int grd = (int)((n + blk - 1) / blk);
        edge_max_kernel<<<grd, blk, 0, stream>>>(src_a, dst_a, el_a, er_a, m_a);
        edge_max_kernel<<<grd, blk, 0, stream>>>(src_b, dst_b, el_b, er_b, m_b);
    }

    // 4) edge softmax + message scatter (dominant, memory/atomic bound)
    {
        const int grd = (NEDGE + 7) / 8;          // 8 edges (waves) per block
        edge_scatter_kernel<<<grd, 256, 0, stream>>>(src_a, dst_a, el_a, er_a, m_a, feat_a, den_a, acc_a);
        edge_scatter_kernel<<<grd, 256, 0, stream>>>(src_b, dst_b, el_b, er_b, m_b, feat_b, den_b, acc_b);
    }

    // 5) normalize + bias + ELU (acc -> z, in place)
    {
        const size_t n = (size_t)NODES * FDIM; const int blk = 256;
        const int grd = (int)((n + blk - 1) / blk);
        finalize_z_kernel<<<grd, blk, 0, stream>>>(den_a, bias_a, acc_a);
        finalize_z_kernel<<<grd, blk, 0, stream>>>(den_b, bias_b, acc_b);
    }

    // 6) semantic attention logits (WMMA GEMM + fused tanh epilogue)
    sem_q_kernel<<<dim3(NODES / 16, 1), wblk, 0, stream>>>(acc_a, Wsem, bsem, wsem, q_a);
    sem_q_kernel<<<dim3(NODES / 16, 1), wblk, 0, stream>>>(acc_b, Wsem, bsem, wsem, q_b);

    // 7) global mean of q per metapath
    qsum_kernel<<<(NODES + 255) / 256, 256, 0, stream>>>(q_a, q_b, sums);

    // 8) softmax(beta) and weighted combine
    {
        const size_t n = (size_t)NODES * FDIM; const int blk = 256;
        const int grd = (int)((n + blk - 1) / blk);
        combine_kernel<<<grd, blk, 0, stream>>>(acc_a, acc_b, sums, out);
    }
}